// GraphAttention_26319559590120
// MI455X (gfx1250) — compile-verified
//
#include <hip/hip_runtime.h>
#include <hip/hip_bf16.h>
#include <cstdint>
#include <cstddef>

typedef __attribute__((ext_vector_type(2))) float v2f;
typedef __attribute__((ext_vector_type(8))) float v8f;

#define C_DIM   128
#define A_ALL_D 256
#define RAD_IN_D 32
#define FC1_D    64
#define FC2_D    64
#define H_D       8
#define AH_D     16
#define VH_D     16

__device__ __forceinline__ float sigmoidf_(float x) { return 1.0f / (1.0f + __expf(-x)); }

// float atomic max via the standard signed-max / unsigned-min bit trick
__device__ __forceinline__ void atomicMaxF32(float* addr, float v) {
    int vi = __float_as_int(v);
    if (vi >= 0) atomicMax((int*)addr, vi);
    else         atomicMin((unsigned int*)addr, (unsigned int)vi);
}

// Direct async global->LDS 16B copy (CDNA5 GLOBAL_LOAD_ASYNC_TO_LDS_B128, ASYNCcnt).
// LDS aperture keeps the wave-relative LDS offset in addr[31:0], so truncating the
// flat pointer yields the VDST byte address the instruction expects.
__device__ __forceinline__ void async_g2l_b128(const void* gptr, void* lds_ptr) {
    unsigned int laddr = (unsigned int)(uintptr_t)lds_ptr;
    unsigned long long gaddr = (unsigned long long)(uintptr_t)gptr;
    asm volatile("global_load_async_to_lds_b128 %0, %1, off"
                 :
                 : "v"(laddr), "v"(gaddr)
                 : "memory");
}
__device__ __forceinline__ void wait_async_all() {
    asm volatile("s_wait_asynccnt 0" ::: "memory");
}

// One 16x16 f32 output tile: A (16xK, row-major, LDS) x B (Kx16 slice of row-major KxNB global)
// using V_WMMA_F32_16X16X4_F32. Operand layouts per CDNA5 ISA 7.12.2:
//   A: lanes 0-15 & 16-31 both hold M=lane&15; VGPR0/1 = K {0,1} (low half) / {2,3} (high half)
//   B: N = lane&15; VGPR0/1 = K {0,1} (low half) / {2,3} (high half)
//   C/D: VGPR g -> M = g + 8*(lane>>4), N = lane&15
template <int K>
__device__ __forceinline__ v8f wmma_tile(const float* __restrict__ As, int lda,
                                         const float* __restrict__ Bg, int ldb,
                                         int lane, v8f acc) {
    const int half = lane >> 4;
    const int l16  = lane & 15;
    #pragma unroll
    for (int k = 0; k < K; k += 4) {
        v2f a, b;
        a.x = As[l16 * lda + k + 2 * half + 0];
        a.y = As[l16 * lda + k + 2 * half + 1];
        b.x = Bg[(size_t)(k + 2 * half + 0) * ldb + l16];
        b.y = Bg[(size_t)(k + 2 * half + 1) * ldb + l16];
        acc = __builtin_amdgcn_wmma_f32_16x16x4_f32(false, a, false, b, (short)0, acc,
                                                    false, false);
    }
    return acc;
}

// ---------------------------------------------------------------- init
__global__ void init_kernel(float* __restrict__ nodemax, float* __restrict__ nodesum,
                            float* __restrict__ outacc, int N) {
    int t = blockIdx.x * blockDim.x + threadIdx.x;
    if (t < N * H_D) { nodemax[t] = -__builtin_huge_valf(); nodesum[t] = 0.0f; }
    if (t < N * C_DIM) outacc[t] = 0.0f;
}

// ------------------------------------------------- node GEMMs (msg_src/msg_dst)
__global__ __launch_bounds__(256) void node_linear_kernel(
    const float* __restrict__ x, const float* __restrict__ w_src,
    const float* __restrict__ b_src, const float* __restrict__ w_dst,
    float* __restrict__ msg_src, float* __restrict__ msg_dst, int N) {
    __shared__ float Xs[16][C_DIM + 4];
    const int tid  = threadIdx.x;
    const int wave = tid >> 5, lane = tid & 31;
    const int half = lane >> 4, l16 = lane & 15;
    const int r0   = blockIdx.x * 16;

    // async-stage the 16x128 input tile straight into (padded) LDS
    for (int q = tid; q < 16 * 32; q += 256) {
        int row = q >> 5, c4 = q & 31;
        int rr = min(r0 + row, N - 1);
        async_g2l_b128(x + (size_t)rr * C_DIM + c4 * 4, &Xs[row][c4 * 4]);
    }
    wait_async_all();
    __syncthreads();

    #pragma unroll
    for (int mat = 0; mat < 2; ++mat) {
        const float* W = mat ? w_dst : w_src;
        v8f acc = {};
        acc = wmma_tile<C_DIM>(&Xs[0][0], C_DIM + 4, W + wave * 16, C_DIM, lane, acc);
        float bias = mat ? 0.0f : b_src[wave * 16 + l16];
        float* out = mat ? msg_dst : msg_src;
        #pragma unroll
        for (int g = 0; g < 8; ++g) {
            int row = g + 8 * half;
            if (r0 + row < N)
                out[(size_t)(r0 + row) * C_DIM + wave * 16 + l16] = acc[g] + bias;
        }
    }
}

// ---------------- fused edge pipeline: gather + radial MLP + sep GEMM + scores
__global__ __launch_bounds__(256) void edge_kernel(
    const float* __restrict__ msg_src, const float* __restrict__ msg_dst,
    const int* __restrict__ edge_src, const int* __restrict__ edge_dst,
    const float* __restrict__ edge_attr, const float* __restrict__ edge_scalars,
    const float* __restrict__ fc_w1, const float* __restrict__ fc_b1,
    const float* __restrict__ fc_w2, const float* __restrict__ fc_b2,
    const float* __restrict__ fc_w3, const float* __restrict__ fc_b3,
    const float* __restrict__ w_sep, const float* __restrict__ b_sep,
    const float* __restrict__ alpha_dot,
    float* __restrict__ score_out, float* __restrict__ nodemax,
    __hip_bfloat16* __restrict__ value_out, int E) {
    __shared__ float Ms[16][C_DIM + 4];
    __shared__ float Ss[16][RAD_IN_D + 4];
    __shared__ float H1s[16][FC1_D + 4];
    __shared__ float H2s[16][FC2_D + 4];
    __shared__ int   esrc[16], edst_s[16];
    __shared__ float eattr[16];

    const int tid  = threadIdx.x;
    const int wave = tid >> 5, lane = tid & 31;
    const int half = lane >> 4, l16 = lane & 15;
    const int e0   = blockIdx.x * 16;

    if (tid < 16) {
        int e = min(e0 + tid, E - 1);
        esrc[tid]   = edge_src[e];
        edst_s[tid] = edge_dst[e];
        eattr[tid]  = edge_attr[e];
    }
    __syncthreads();

    // async-stage edge_scalars tile (16 x 32) -> overlaps with the gather below
    for (int q = tid; q < 16 * 8; q += 256) {
        int row = q >> 3, c4 = q & 7;
        int e = min(e0 + row, E - 1);
        async_g2l_b128(edge_scalars + (size_t)e * RAD_IN_D + c4 * 4, &Ss[row][c4 * 4]);
    }
    // gather message tile (16 x 128): msg_src[src] + msg_dst[dst] (needs VALU add)
    for (int q = tid; q < 16 * 32; q += 256) {
        int row = q >> 5, c4 = q & 31;
        float4 a = ((const float4*)(msg_src + (size_t)esrc[row] * C_DIM))[c4];
        float4 b = ((const float4*)(msg_dst + (size_t)edst_s[row] * C_DIM))[c4];
        Ms[row][c4 * 4 + 0] = a.x + b.x; Ms[row][c4 * 4 + 1] = a.y + b.y;
        Ms[row][c4 * 4 + 2] = a.z + b.z; Ms[row][c4 * 4 + 3] = a.w + b.w;
    }
    wait_async_all();
    __syncthreads();

    // radial MLP layer 1: (16x32)@(32x64) + silu
    if (wave < 4) {
        v8f acc = {};
        acc = wmma_tile<RAD_IN_D>(&Ss[0][0], RAD_IN_D + 4, fc_w1 + wave * 16, FC1_D, lane, acc);
        float b1 = fc_b1[wave * 16 + l16];
        #pragma unroll
        for (int g = 0; g < 8; ++g) {
            float xv = acc[g] + b1;
            H1s[g + 8 * half][wave * 16 + l16] = xv * sigmoidf_(xv);
        }
    }
    __syncthreads();
    // layer 2: (16x64)@(64x64) + silu
    if (wave < 4) {
        v8f acc = {};
        acc = wmma_tile<FC1_D>(&H1s[0][0], FC1_D + 4, fc_w2 + wave * 16, FC2_D, lane, acc);
        float b2 = fc_b2[wave * 16 + l16];
        #pragma unroll
        for (int g = 0; g < 8; ++g) {
            float xv = acc[g] + b2;
            H2s[g + 8 * half][wave * 16 + l16] = xv * sigmoidf_(xv);
        }
    }
    __syncthreads();
    // layer 3: (16x64)@(64x128); fuse  M *= edge_attr * rad_w
    {
        v8f acc = {};
        acc = wmma_tile<FC2_D>(&H2s[0][0], FC2_D + 4, fc_w3 + wave * 16, C_DIM, lane, acc);
        float b3 = fc_b3[wave * 16 + l16];
        #pragma unroll
        for (int g = 0; g < 8; ++g) {
            int row = g + 8 * half;
            float r = acc[g] + b3;
            Ms[row][wave * 16 + l16] *= eattr[row] * r;
        }
    }
    __syncthreads();

    // separable GEMM (16x128)@(128x256); tt even -> alpha block of head tt/2, odd -> value
    #pragma unroll
    for (int i = 0; i < 2; ++i) {
        int tt = wave * 2 + i;
        int h  = tt >> 1;
        v8f acc = {};
        acc = wmma_tile<C_DIM>(&Ms[0][0], C_DIM + 4, w_sep + tt * 16, A_ALL_D, lane, acc);
        float bias = b_sep[tt * 16 + l16];
        if ((tt & 1) == 0) {
            float ad = alpha_dot[h * AH_D + l16];
            #pragma unroll
            for (int g = 0; g < 8; ++g) {
                float xv = acc[g] + bias;
                float sg = sigmoidf_(xv);
                float sl = 0.6f * xv + 0.4f * xv * (2.0f * sg - 1.0f); // smooth leaky relu, a=0.2
                float p  = sl * ad;
                p += __shfl_xor(p, 1);
                p += __shfl_xor(p, 2);
                p += __shfl_xor(p, 4);
                p += __shfl_xor(p, 8);
                if (l16 == 0) {
                    int row = g + 8 * half;
                    int e   = e0 + row;
                    if (e < E) {
                        score_out[(size_t)e * H_D + h] = p;
                        atomicMaxF32(&nodemax[(size_t)edst_s[row] * H_D + h], p);
                    }
                }
            }
        } else {
            #pragma unroll
            for (int g = 0; g < 8; ++g) {
                int row = g + 8 * half;
                int e   = e0 + row;
                if (e < E)
                    value_out[(size_t)e * (H_D * VH_D) + h * VH_D + l16] =
                        __float2bfloat16(acc[g] + bias);
            }
        }
    }
}

// ------------------------------------------- segment softmax denominator
__global__ void softmax_sum_kernel(const int* __restrict__ edge_dst,
                                   float* __restrict__ score,
                                   const float* __restrict__ nodemax,
                                   float* __restrict__ nodesum, int E) {
    int t = blockIdx.x * blockDim.x + threadIdx.x;
    if (t >= E * H_D) return;
    int e = t >> 3, h = t & 7;
    int d = edge_dst[e];
    float ex = __expf(score[t] - nodemax[d * H_D + h]);
    score[t] = ex; // cache numerator
    atomicAdd(&nodesum[d * H_D + h], ex);
}

// ------------------------------------------- attention-weighted scatter sum
__global__ void scatter_kernel(const int* __restrict__ edge_dst,
                               const float* __restrict__ score,
                               const float* __restrict__ nodesum,
                               const __hip_bfloat16* __restrict__ value,
                               float* __restrict__ outacc, int E) {
    size_t t = (size_t)blockIdx.x * blockDim.x + threadIdx.x;
    if (t >= (size_t)E * C_DIM) return;
    int e = (int)(t >> 7);
    int c = (int)(t & 127);
    int h = c >> 4;
    int d = edge_dst[e];
    float att = score[(size_t)e * H_D + h] / (nodesum[d * H_D + h] + 1e-16f);
    float v = __bfloat162float(value[t]);
    atomicAdd(&outacc[(size_t)d * C_DIM + c], v * att);
}

// ------------------------------------------- final projection
__global__ __launch_bounds__(256) void proj_kernel(const float* __restrict__ outacc,
                                                   const float* __restrict__ w_proj,
                                                   const float* __restrict__ b_proj,
                                                   float* __restrict__ out, int N) {
    __shared__ float Xs[16][C_DIM + 4];
    const int tid  = threadIdx.x;
    const int wave = tid >> 5, lane = tid & 31;
    const int half = lane >> 4, l16 = lane & 15;
    const int r0   = blockIdx.x * 16;

    for (int q = tid; q < 16 * 32; q += 256) {
        int row = q >> 5, c4 = q & 31;
        int rr = min(r0 + row, N - 1);
        async_g2l_b128(outacc + (size_t)rr * C_DIM + c4 * 4, &Xs[row][c4 * 4]);
    }
    wait_async_all();
    __syncthreads();

    v8f acc = {};
    acc = wmma_tile<C_DIM>(&Xs[0][0], C_DIM + 4, w_proj + wave * 16, C_DIM, lane, acc);
    float bias = b_proj[wave * 16 + l16];
    #pragma unroll
    for (int g = 0; g < 8; ++g) {
        int row = g + 8 * half;
        if (r0 + row < N)
            out[(size_t)(r0 + row) * C_DIM + wave * 16 + l16] = acc[g] + bias;
    }
}

extern "C" void kernel_launch(void* const* d_in, const int* in_sizes, int n_in,
                              void* d_out, int out_size, void* d_ws, size_t ws_size,
                              hipStream_t stream) {
    (void)n_in; (void)out_size; (void)ws_size;
    const float* node_input   = (const float*)d_in[0];
    const float* edge_attr    = (const float*)d_in[2];
    const float* edge_scalars = (const float*)d_in[3];
    const int*   edge_src     = (const int*)d_in[4];
    const int*   edge_dst     = (const int*)d_in[5];
    const float* w_src  = (const float*)d_in[7];
    const float* b_src  = (const float*)d_in[8];
    const float* w_dst  = (const float*)d_in[9];
    const float* fc_w1  = (const float*)d_in[10];
    const float* fc_b1  = (const float*)d_in[11];
    const float* fc_w2  = (const float*)d_in[12];
    const float* fc_b2  = (const float*)d_in[13];
    const float* fc_w3  = (const float*)d_in[14];
    const float* fc_b3  = (const float*)d_in[15];
    const float* w_sep  = (const float*)d_in[16];
    const float* b_sep  = (const float*)d_in[17];
    const float* alpha_dot = (const float*)d_in[18];
    const float* w_proj = (const float*)d_in[19];
    const float* b_proj = (const float*)d_in[20];

    const int N = in_sizes[0] / C_DIM;
    const int E = in_sizes[4];

    char* ws = (char*)d_ws;
    size_t off = 0;
    auto alloc = [&](size_t bytes) -> void* {
        void* p = ws + off;
        off += (bytes + 255) & ~(size_t)255;
        return p;
    };
    float* msg_src = (float*)alloc((size_t)N * C_DIM * sizeof(float));
    float* msg_dst = (float*)alloc((size_t)N * C_DIM * sizeof(float));
    float* score   = (float*)alloc((size_t)E * H_D * sizeof(float));
    float* nodemax = (float*)alloc((size_t)N * H_D * sizeof(float));
    float* nodesum = (float*)alloc((size_t)N * H_D * sizeof(float));
    float* outacc  = (float*)alloc((size_t)N * C_DIM * sizeof(float));
    __hip_bfloat16* value = (__hip_bfloat16*)alloc((size_t)E * C_DIM * sizeof(__hip_bfloat16));

    init_kernel<<<(N * C_DIM + 255) / 256, 256, 0, stream>>>(nodemax, nodesum, outacc, N);
    node_linear_kernel<<<(N + 15) / 16, 256, 0, stream>>>(node_input, w_src, b_src, w_dst,
                                                          msg_src, msg_dst, N);
    edge_kernel<<<(E + 15) / 16, 256, 0, stream>>>(msg_src, msg_dst, edge_src, edge_dst,
                                                   edge_attr, edge_scalars,
                                                   fc_w1, fc_b1, fc_w2, fc_b2, fc_w3, fc_b3,
                                                   w_sep, b_sep, alpha_dot,
                                                   score, nodemax, value, E);
    softmax_sum_kernel<<<(E * H_D + 255) / 256, 256, 0, stream>>>(edge_dst, score, nodemax,
                                                                  nodesum, E);
    scatter_kernel<<<(int)(((size_t)E * C_DIM + 255) / 256), 256, 0, stream>>>(
        edge_dst, score, nodesum, value, outacc, E);
    proj_kernel<<<(N + 15) / 16, 256, 0, stream>>>(outacc, w_proj, b_proj, (float*)d_out, N);
}